// BirdModel_10325101379669
// MI455X (gfx1250) — compile-verified
//
#include <hip/hip_runtime.h>

#define DEVFN __device__ __forceinline__

typedef __attribute__((ext_vector_type(2))) float v2f;
typedef __attribute__((ext_vector_type(8))) float v8f;

// ---------------- problem sizes (fixed by reference) ----------------
constexpr int B_   = 256;
constexpr int T_   = 1024;
constexpr int IN_  = 128;
constexpr int H_   = 64;
constexpr int G4_  = 4 * H_;   // 256 gate columns
constexpr int OUT_ = 100;

// ---------------- tiling ----------------
constexpr int ROWS   = 16;              // batch rows per block (WMMA M)
constexpr int NWAVE  = 4;               // waves per block; wave w owns h cols [16w,16w+16)
constexpr int TPB    = NWAVE * 32;      // 128 threads (wave32)
constexpr int NBLK   = B_ / ROWS;       // 16 scan blocks
constexpr int TCHUNK = 64;              // timesteps per phase-1 block

// ---------------- LDS layout ----------------
// Weights are staged PRE-SWIZZLED into B-fragment layout:
//   wf[((kchunk*2 + hh) * WFS + n)] is a float2 = { W[n][4*kchunk+2*hh],
//                                                   W[n][4*kchunk+2*hh+1] }
// so one aligned ds_load_b64 fills a WMMA B operand register pair directly.
// WFS padded so hh=1 rows land 32 banks away from hh=0 rows (conflict-free).
constexpr int WFS = G4_ + 16;                       // float2 stride = 272
constexpr int WIH_DW = (IN_ / 4) * 2 * WFS * 2;     // 34816 dwords
constexpr int WHH_DW = (H_  / 4) * 2 * WFS * 2;     // 17408 dwords
constexpr int HS = H_ + 4;    // 68 : h row stride (= 4 mod 64 -> conflict-free b64 A-frags)
constexpr int XS = IN_ + 4;   // 132: staged specs-tile row stride (= 4 mod 64)

// ---------------- WMMA: fp32 16x16x4, D = A*B + C ----------------
DEVFN v8f wmma_f32(v2f a, v2f b, v8f c) {
  // (neg_a, A, neg_b, B, c_mod, C, reuse_a, reuse_b)
  return __builtin_amdgcn_wmma_f32_16x16x4_f32(false, a, false, b, (short)0, c,
                                               false, false);
}

// A-fragment from row-major LDS [M][stride]: lane (ln,hh) reg r <-> (m=ln, k=kk+2*hh+r)
DEVFN v2f ldA(const float* base, int stride, int ln, int kk, int hh) {
  return *(const v2f*)(base + ln * stride + kk + 2 * hh);   // one ds_load_b64
}

// B-fragment from fragment-swizzled weight LDS: single aligned ds_load_b64.
DEVFN v2f ldB(const float* wf, int nt, int kk, int ln, int hh) {
  return *(const v2f*)(wf + (((kk >> 2) * 2 + hh) * WFS + nt * 16 + ln) * 2);
}

// Stage W[g][k] (row-major [G4][K]) into the fragment-swizzled LDS layout.
DEVFN void stage_wt(float* dst, const float* __restrict__ src, int K, int tid) {
  for (int idx = tid; idx < G4_ * K; idx += TPB) {
    const int g = idx / K, k = idx - g * K;        // coalesced global read
    const int kc = k >> 2, hh = (k >> 1) & 1, r = k & 1;
    dst[((kc * 2 + hh) * WFS + g) * 2 + r] = src[idx];
  }
}

// Cooperative 16x128 specs tile: 8 threads/row, 16 floats (4x float4) each
struct XPre { float4 v[4]; };
DEVFN XPre load_x(const float* __restrict__ specs, int b0, int t, int tid) {
  XPre r;
  const int rr = tid >> 3, cs = (tid & 7) * 16;
  const float* p = specs + ((size_t)(b0 + rr) * T_ + t) * IN_ + cs;
  #pragma unroll
  for (int j = 0; j < 4; ++j) r.v[j] = ((const float4*)p)[j];
  return r;
}
DEVFN void store_x(float* xs, const XPre& r, int tid) {
  const int rr = tid >> 3, cs = (tid & 7) * 16;
  float* p = xs + rr * XS + cs;
  #pragma unroll
  for (int j = 0; j < 4; ++j) ((float4*)p)[j] = r.v[j];
}

DEVFN float sigm(float x) { return 1.0f / (1.0f + __expf(-x)); }
DEVFN float tanh_fast(float x) {
  x = fminf(15.0f, fmaxf(-15.0f, x));
  const float e = __expf(2.0f * x);
  return (e - 1.0f) / (e + 1.0f);
}

// =====================================================================
// Phase 1: x_proj[t][b][g] = specs[b][t][:] . W_ih[g][:] + b_ih[g] + b_hh[g]
// grid = (B/16, T/TCHUNK), 128 threads. Fully parallel WMMA GEMM.
// =====================================================================
__global__ __launch_bounds__(TPB) void lstm_xproj_kernel(
    const float* __restrict__ specs, const float* __restrict__ W_ih,
    const float* __restrict__ b_ih, const float* __restrict__ b_hh,
    float* __restrict__ xp)
{
  extern __shared__ float smem[];
  float* wih_s = smem;                    // WIH_DW
  float* xs    = smem + WIH_DW;           // 2 * ROWS * XS (double buffer)

  const int tid = threadIdx.x;
  const int b0  = blockIdx.x * ROWS;
  const int t0  = blockIdx.y * TCHUNK;
  const int lane = tid & 31, wv = tid >> 5, ln = lane & 15, hh = lane >> 4;

  stage_wt(wih_s, W_ih, IN_, tid);

  float bias[4];
  #pragma unroll
  for (int g = 0; g < 4; ++g) {
    const int n = (wv + 4 * g) * 16 + ln;
    bias[g] = b_ih[n] + b_hh[n];
  }

  { XPre p0 = load_x(specs, b0, t0, tid); store_x(xs, p0, tid); }
  __syncthreads();

  for (int tt = 0; tt < TCHUNK; ++tt) {
    const int t = t0 + tt;
    float* xcur = xs + (tt & 1) * ROWS * XS;
    float* xnxt = xs + ((tt + 1) & 1) * ROWS * XS;
    const bool more = (tt + 1) < TCHUNK;
    XPre pre;
    if (more) pre = load_x(specs, b0, t + 1, tid);   // overlap with WMMAs

    v8f acc[4];
    #pragma unroll
    for (int g = 0; g < 4; ++g)
      #pragma unroll
      for (int v = 0; v < 8; ++v) acc[g][v] = bias[g];

    #pragma unroll 8
    for (int kk = 0; kk < IN_; kk += 4) {
      const v2f a = ldA(xcur, XS, ln, kk, hh);       // shared across the 4 n-tiles
      #pragma unroll
      for (int g = 0; g < 4; ++g)
        acc[g] = wmma_f32(a, ldB(wih_s, wv + 4 * g, kk, ln, hh), acc[g]);
    }

    #pragma unroll
    for (int g = 0; g < 4; ++g) {
      const int n = (wv + 4 * g) * 16 + ln;
      #pragma unroll
      for (int v = 0; v < 8; ++v)
        xp[((size_t)t * B_ + b0 + v + 8 * hh) * G4_ + n] = acc[g][v];
    }

    if (more) store_x(xnxt, pre, tid);
    __syncthreads();
  }
}

// =====================================================================
// Phase 2: sequential LSTM scan. grid = 16 blocks x 128 threads.
// PRECOMP: gates from x_proj in d_ws (loads overlapped with WMMA chain).
// FUSED:   input projection recomputed in-loop from LDS-staged specs tile.
// Epilogue: out = h_T @ W_out^T + b_out (h already in LDS).
// =====================================================================
template <bool PRECOMP>
__global__ __launch_bounds__(TPB) void lstm_scan_kernel(
    const float* __restrict__ specs, const float* __restrict__ xp,
    const float* __restrict__ W_ih,  const float* __restrict__ W_hh,
    const float* __restrict__ b_ih,  const float* __restrict__ b_hh,
    const float* __restrict__ W_out, const float* __restrict__ b_out,
    float* __restrict__ out)
{
  extern __shared__ float smem[];
  float* whh_s = smem;                                   // WHH_DW
  float* hbuf  = whh_s + WHH_DW;                         // ROWS * HS
  float* wih_s = hbuf + ROWS * HS;                       // FUSED only: WIH_DW
  float* xs    = wih_s + (PRECOMP ? 0 : WIH_DW);         // FUSED only: 2*ROWS*XS

  const int tid = threadIdx.x;
  const int b0  = blockIdx.x * ROWS;
  const int lane = tid & 31, wv = tid >> 5, ln = lane & 15, hh = lane >> 4;

  stage_wt(whh_s, W_hh, H_, tid);
  if (!PRECOMP) stage_wt(wih_s, W_ih, IN_, tid);
  for (int idx = tid; idx < ROWS * HS; idx += TPB) hbuf[idx] = 0.0f;  // h0 = 0

  float bias[4];
  if (!PRECOMP) {
    #pragma unroll
    for (int g = 0; g < 4; ++g) {
      const int n = (wv + 4 * g) * 16 + ln;
      bias[g] = b_ih[n] + b_hh[n];
    }
  }

  v8f cst;                                               // c state: registers only
  #pragma unroll
  for (int v = 0; v < 8; ++v) cst[v] = 0.0f;

  if (!PRECOMP) { XPre p0 = load_x(specs, b0, 0, tid); store_x(xs, p0, tid); }
  __syncthreads();

  for (int t = 0; t < T_; ++t) {
    float* xcur = xs + (t & 1) * ROWS * XS;
    float* xnxt = xs + ((t + 1) & 1) * ROWS * XS;
    const bool more = (t + 1) < T_;
    XPre pre;
    if (!PRECOMP && more) {
      pre = load_x(specs, b0, t + 1, tid);
      if (t + 2 < T_) {
        const int rr = tid >> 3, cs = (tid & 7) * 16;
        __builtin_prefetch(specs + ((size_t)(b0 + rr) * T_ + (t + 2)) * IN_ + cs, 0, 0);
      }
    }

    // ---- gates: acc[g] covers n-tile (wv + 4g): g=0:i, 1:f, 2:g, 3:o ----
    v8f acc[4];
    float xf[4][8];                      // PRECOMP: x_proj, loads overlap WMMAs
    if (PRECOMP) {
      const float* xprow = xp + (size_t)t * B_ * G4_;
      #pragma unroll
      for (int g = 0; g < 4; ++g) {
        const int n = (wv + 4 * g) * 16 + ln;
        #pragma unroll
        for (int v = 0; v < 8; ++v)
          xf[g][v] = xprow[(size_t)(b0 + v + 8 * hh) * G4_ + n];
      }
      #pragma unroll
      for (int g = 0; g < 4; ++g)
        #pragma unroll
        for (int v = 0; v < 8; ++v) acc[g][v] = 0.0f;
    } else {
      #pragma unroll
      for (int g = 0; g < 4; ++g)
        #pragma unroll
        for (int v = 0; v < 8; ++v) acc[g][v] = bias[g];
      #pragma unroll 8
      for (int kk = 0; kk < IN_; kk += 4) {
        const v2f a = ldA(xcur, XS, ln, kk, hh);
        #pragma unroll
        for (int g = 0; g < 4; ++g)
          acc[g] = wmma_f32(a, ldB(wih_s, wv + 4 * g, kk, ln, hh), acc[g]);
      }
    }
    // recurrent h @ W_hh^T (K = 64 -> 16 chained WMMAs x 4 independent chains)
    #pragma unroll
    for (int kk = 0; kk < H_; kk += 4) {
      const v2f a = ldA(hbuf, HS, ln, kk, hh);
      #pragma unroll
      for (int g = 0; g < 4; ++g)
        acc[g] = wmma_f32(a, ldB(whh_s, wv + 4 * g, kk, ln, hh), acc[g]);
    }
    if (PRECOMP) {                       // fold in x_proj after the chain
      #pragma unroll
      for (int g = 0; g < 4; ++g)
        #pragma unroll
        for (int v = 0; v < 8; ++v) acc[g][v] += xf[g][v];
    }

    // ---- elementwise cell update (c in registers) ----
    float hv[8];
    #pragma unroll
    for (int v = 0; v < 8; ++v) {
      const float ig = sigm(acc[0][v]);
      const float fg = sigm(acc[1][v]);
      const float gg = tanh_fast(acc[2][v]);
      const float og = sigm(acc[3][v]);
      const float c  = fg * cst[v] + ig * gg;
      cst[v] = c;
      hv[v]  = og * tanh_fast(c);
    }

    __syncthreads();                                   // all reads of hbuf done
    #pragma unroll
    for (int v = 0; v < 8; ++v)
      hbuf[(v + 8 * hh) * HS + wv * 16 + ln] = hv[v];  // conflict-free banks
    if (!PRECOMP && more) store_x(xnxt, pre, tid);
    __syncthreads();                                   // new h / x tile visible
  }

  // ---- epilogue: out[b0+r][o] = h[r][:] . W_out[o][:] + b_out[o] ----
  for (int idx = tid; idx < ROWS * OUT_; idx += TPB) {
    const int r = idx / OUT_, o = idx - r * OUT_;
    const float* hr = hbuf + r * HS;
    const float* wo = W_out + o * H_;
    float s = b_out[o];
    #pragma unroll
    for (int k = 0; k < H_; ++k) s += hr[k] * wo[k];
    out[(size_t)(b0 + r) * OUT_ + o] = s;
  }
}

// =====================================================================
extern "C" void kernel_launch(void* const* d_in, const int* in_sizes, int n_in,
                              void* d_out, int out_size, void* d_ws, size_t ws_size,
                              hipStream_t stream) {
  const float* specs = (const float*)d_in[0];
  const float* W_ih  = (const float*)d_in[1];
  const float* W_hh  = (const float*)d_in[2];
  const float* b_ih  = (const float*)d_in[3];
  const float* b_hh  = (const float*)d_in[4];
  const float* W_out = (const float*)d_in[5];
  const float* b_out = (const float*)d_in[6];
  float* out = (float*)d_out;

  const size_t xp_bytes = (size_t)T_ * B_ * G4_ * sizeof(float);  // 256 MB
  if (ws_size >= xp_bytes) {
    float* xp = (float*)d_ws;
    const size_t lds1 = (size_t)(WIH_DW + 2 * ROWS * XS) * sizeof(float);
    lstm_xproj_kernel<<<dim3(B_ / ROWS, T_ / TCHUNK), TPB, lds1, stream>>>(
        specs, W_ih, b_ih, b_hh, xp);
    const size_t lds2 = (size_t)(WHH_DW + ROWS * HS) * sizeof(float);
    lstm_scan_kernel<true><<<dim3(NBLK), TPB, lds2, stream>>>(
        specs, xp, W_ih, W_hh, b_ih, b_hh, W_out, b_out, out);
  } else {
    const size_t lds = (size_t)(WHH_DW + ROWS * HS + WIH_DW + 2 * ROWS * XS) *
                       sizeof(float);
    lstm_scan_kernel<false><<<dim3(NBLK), TPB, lds, stream>>>(
        specs, nullptr, W_ih, W_hh, b_ih, b_hh, W_out, b_out, out);
  }
}